// ArcFaceLoss_52029233824318
// MI455X (gfx1250) — compile-verified
//
#include <hip/hip_runtime.h>
#include <hip/hip_bf16.h>
#include <cstdint>

#define TPB 256
#define TILE_F4 1024           // float4 per LDS buffer = 16 KB
#define F4_PER_THREAD (TILE_F4 / TPB)   // 4

// Match the builtin's declared parameter types: pointer to 16-byte int vector,
// global (AS1) source and LDS (AS3) destination.
typedef int v4i __attribute__((vector_size(16)));
typedef __attribute__((address_space(1))) v4i gv4i_t;
typedef __attribute__((address_space(3))) v4i lv4i_t;

// ---- CDNA5 async global->LDS load (B128), ASYNCcnt-tracked ----
__device__ __forceinline__ void async_load_f4(const float4* g, float4* l) {
#if __has_builtin(__builtin_amdgcn_global_load_async_to_lds_b128)
    __builtin_amdgcn_global_load_async_to_lds_b128(
        (gv4i_t*)g, (lv4i_t*)l, 0, 0);
#else
    uint32_t lds_addr = (uint32_t)(uintptr_t)l;   // low 32 bits of generic = LDS byte offset
    asm volatile("global_load_async_to_lds_b128 %0, %1, off"
                 :: "v"(lds_addr), "v"(g) : "memory");
#endif
}

__device__ __forceinline__ void wait_async_le(int n /*compile-time 0 or 4*/) {
#if __has_builtin(__builtin_amdgcn_s_wait_asynccnt)
    if (n == 0) __builtin_amdgcn_s_wait_asynccnt(0);
    else        __builtin_amdgcn_s_wait_asynccnt(4);
#else
    if (n == 0) asm volatile("s_wait_asynccnt 0" ::: "memory");
    else        asm volatile("s_wait_asynccnt 4" ::: "memory");
#endif
}

// ---- fast transcendentals (v_exp_f32 / v_log_f32 / v_sqrt_f32) ----
__device__ __forceinline__ float fast_exp2(float x) {
#if __has_builtin(__builtin_amdgcn_exp2f)
    return __builtin_amdgcn_exp2f(x);
#else
    return exp2f(x);
#endif
}
__device__ __forceinline__ float fast_log2(float x) {
#if __has_builtin(__builtin_amdgcn_logf)
    return __builtin_amdgcn_logf(x);
#else
    return __log2f(x);
#endif
}
__device__ __forceinline__ float fast_sqrt(float x) {
#if __has_builtin(__builtin_amdgcn_sqrtf)
    return __builtin_amdgcn_sqrtf(x);
#else
    return sqrtf(x);
#endif
}

// loss_b = log( sum_c exp(l_c - 64) ) + 64 - l_target   (shift-invariant; |l| < 64.1)
__global__ __launch_bounds__(TPB) void arcface_row_kernel(
    const float* __restrict__ cosine, const int* __restrict__ label,
    float* __restrict__ row_loss, int B, int C) {

    __shared__ float4 tile[2][TILE_F4];
    __shared__ float red_s[TPB / 32];
    __shared__ float red_t[TPB / 32];

    const int row = blockIdx.x;
    const int tid = threadIdx.x;

    const int lab    = label[row];
    const int tcol   = (lab < 0) ? (C + lab) : lab;  // jax take_along_axis wraps negatives
    const bool margin = (lab >= 0);                   // one_hot(-1) == zeros -> no margin

    const float4* rowp = (const float4*)(cosine + (size_t)row * (size_t)C);
    const int nf4    = C >> 2;                        // C % 4 == 0 (100000)
    const int ntiles = (nf4 + TILE_F4 - 1) / TILE_F4;

    // ---- prologue: issue tile 0 (clamped addresses keep ASYNCcnt uniform per wave) ----
    #pragma unroll
    for (int k = 0; k < F4_PER_THREAD; ++k) {
        int g  = k * TPB + tid;
        int gc = g < nf4 ? g : nf4 - 1;
        async_load_f4(rowp + gc, &tile[0][k * TPB + tid]);
    }

    const float A   = 92.33248261689366f;   // 64 * log2(e)
    const float L2E = 1.4426950408889634f;
    const float CM  = 0.99500416527803f;    // cos(0.1)
    const float SM  = 0.09983341664683f;    // sin(0.1)

    float acc0 = 0.f, acc1 = 0.f, acc2 = 0.f, acc3 = 0.f;
    float tlog = -__builtin_inff();

    for (int t = 0; t < ntiles; ++t) {
        const int p = t & 1;
        if (t + 1 < ntiles) {
            const int base = (t + 1) * TILE_F4;
            #pragma unroll
            for (int k = 0; k < F4_PER_THREAD; ++k) {
                int g  = base + k * TPB + tid;
                int gc = g < nf4 ? g : nf4 - 1;
                async_load_f4(rowp + gc, &tile[p ^ 1][k * TPB + tid]);
            }
            wait_async_le(4);   // 4 newly issued outstanding -> tile t complete
        } else {
            wait_async_le(0);
        }

        const int base = t * TILE_F4;
        #pragma unroll
        for (int k = 0; k < F4_PER_THREAD; ++k) {
            const int g = base + k * TPB + tid;
            if (g < nf4) {
                float4 v = tile[p][k * TPB + tid];
                const int colbase = g << 2;
                if ((unsigned)(tcol - colbase) < 4u) {
                    // rare: tile chunk containing the target column
                    float xs[4] = {v.x, v.y, v.z, v.w};
                    #pragma unroll
                    for (int j = 0; j < 4; ++j) {
                        float x = xs[j];
                        float l;
                        if (colbase + j == tcol) {
                            if (margin) {
                                float sn = fast_sqrt(fmaxf(0.f, 1.f - x * x));
                                l = 64.f * (x * CM - sn * SM);
                            } else {
                                l = 64.f * x;
                            }
                            tlog = l;
                        } else {
                            l = 64.f * x;
                        }
                        acc0 += fast_exp2((l - 64.f) * L2E);
                    }
                } else {
                    acc0 += fast_exp2(fmaf(A, v.x, -A));
                    acc1 += fast_exp2(fmaf(A, v.y, -A));
                    acc2 += fast_exp2(fmaf(A, v.z, -A));
                    acc3 += fast_exp2(fmaf(A, v.w, -A));
                }
            }
        }
    }

    float s = (acc0 + acc1) + (acc2 + acc3);
    // wave32 reduction
    #pragma unroll
    for (int off = 16; off > 0; off >>= 1) {
        s    += __shfl_down(s, off, 32);
        tlog  = fmaxf(tlog, __shfl_down(tlog, off, 32));
    }
    const int wave = tid >> 5;
    if ((tid & 31) == 0) { red_s[wave] = s; red_t[wave] = tlog; }
    __syncthreads();
    if (tid == 0) {
        float S = 0.f, T = -__builtin_inff();
        #pragma unroll
        for (int w = 0; w < TPB / 32; ++w) { S += red_s[w]; T = fmaxf(T, red_t[w]); }
        row_loss[row] = fast_log2(S) * 0.6931471805599453f + 64.f - T;
    }
}

// Deterministic mean over B row losses (single block; no float atomics).
__global__ __launch_bounds__(256) void arcface_mean_kernel(
    const float* __restrict__ row_loss, float* __restrict__ out, int B) {
    __shared__ float red[8];
    float s = 0.f;
    for (int i = threadIdx.x; i < B; i += 256) s += row_loss[i];
    #pragma unroll
    for (int off = 16; off > 0; off >>= 1) s += __shfl_down(s, off, 32);
    if ((threadIdx.x & 31) == 0) red[threadIdx.x >> 5] = s;
    __syncthreads();
    if (threadIdx.x == 0) {
        float S = 0.f;
        #pragma unroll
        for (int w = 0; w < 8; ++w) S += red[w];
        out[0] = S / (float)B;
    }
}

extern "C" void kernel_launch(void* const* d_in, const int* in_sizes, int n_in,
                              void* d_out, int out_size, void* d_ws, size_t ws_size,
                              hipStream_t stream) {
    const float* cosine = (const float*)d_in[0];
    const int*   lab    = (const int*)d_in[1];
    const int B = in_sizes[1];
    const int C = (int)((long long)in_sizes[0] / (long long)B);
    float* row_loss = (float*)d_ws;   // B floats of scratch

    arcface_row_kernel<<<dim3(B), dim3(TPB), 0, stream>>>(cosine, lab, row_loss, B, C);
    arcface_mean_kernel<<<dim3(1), dim3(256), 0, stream>>>(row_loss, (float*)d_out, B);
}